// NESLinear_62517543960898
// MI455X (gfx1250) — compile-verified
//
#include <hip/hip_runtime.h>

typedef __attribute__((ext_vector_type(2))) float v2f;
typedef __attribute__((ext_vector_type(4))) float v4f;
typedef __attribute__((ext_vector_type(8))) float v8f;

#define P_DIM 1024
#define D_DIM 512
#define O_DIM 512

// ---------------------------------------------------------------------------
// Kernel 1: base = x @ W + bias + bias_eps   via V_WMMA_F32_16X16X4_F32
// One wave (32 threads) per 16x16 output tile. k-loop over D in steps of 4.
//
// f32 WMMA operand layouts (ISA 7.12.2):
//   A 16x4 : lane l, vgpr v ->  M = l%16, K = 2*(l/16) + v   (float2 load, K contiguous)
//   B 4x16 : lane l, vgpr v ->  N = l%16, K = 2*(l/16) + v   (two dword loads, stride O)
//   C/D    : lane l, vgpr v ->  N = l%16, M = 8*(l/16) + v
// ---------------------------------------------------------------------------
__global__ __launch_bounds__(32) void nes_base_wmma(
    const float* __restrict__ x,        // [P, D]
    const float* __restrict__ W,        // [D, O]
    const float* __restrict__ bias,     // [O]
    const float* __restrict__ bias_eps, // [P, O]
    float* __restrict__ out)            // [P, O]
{
  const int lane = threadIdx.x & 31;
  const int half = lane >> 4;   // 0 or 1
  const int lm   = lane & 15;
  const int n0   = blockIdx.x * 16;   // output-column tile
  const int m0   = blockIdx.y * 16;   // output-row (p) tile

  v8f acc = {0.f, 0.f, 0.f, 0.f, 0.f, 0.f, 0.f, 0.f};

  const float* arow = x + (size_t)(m0 + lm) * D_DIM;
  for (int k = 0; k < D_DIM; k += 4) {
    const int ka = k + 2 * half;                 // this lane's first K
    v2f a = *(const v2f*)(arow + ka);            // A[M, ka], A[M, ka+1]
    v2f b;
    b.x = W[(size_t)(ka + 0) * O_DIM + n0 + lm]; // B[ka,   N]
    b.y = W[(size_t)(ka + 1) * O_DIM + n0 + lm]; // B[ka+1, N]
    acc = __builtin_amdgcn_wmma_f32_16x16x4_f32(
        /*neg_a=*/false, a, /*neg_b=*/false, b,
        /*c_mod=*/(short)0, acc, /*reuse_a=*/false, /*reuse_b=*/false);
  }

  const float bn = bias[n0 + lm];
  #pragma unroll
  for (int v = 0; v < 8; ++v) {
    const int m = m0 + 8 * half + v;
    const size_t idx = (size_t)m * O_DIM + n0 + lm;
    out[idx] = acc[v] + bn + bias_eps[idx];
  }
}

// ---------------------------------------------------------------------------
// Kernel 2: out[p,:] += sum_d x[p,d] * weight_eps[p,d,:]
// One block per p; 128 threads, each owning 4 contiguous output columns.
// weight_eps (1 GiB) is streamed once with non-temporal b128 loads (HBM-bound:
// ~46 us at 23.3 TB/s). x[p,:] staged in LDS, broadcast-read per d.
// ---------------------------------------------------------------------------
__global__ __launch_bounds__(128) void nes_eps_stream(
    const float* __restrict__ x,    // [P, D]
    const float* __restrict__ eps,  // [P, D, O]
    float* __restrict__ out)        // [P, O]
{
  __shared__ float xs[D_DIM];
  const int p = blockIdx.x;
  const int t = threadIdx.x;   // 0..127 -> columns 4t..4t+3

  for (int i = t; i < D_DIM; i += 128)
    xs[i] = x[(size_t)p * D_DIM + i];
  __syncthreads();

  const v4f* ep = (const v4f*)(eps + (size_t)p * D_DIM * O_DIM) + t;
  v4f acc = {0.f, 0.f, 0.f, 0.f};

  for (int d = 0; d < D_DIM; d += 8) {
    // two LDS b128 broadcast loads cover 8 x-values
    v4f xa = *(const v4f*)(xs + d);
    v4f xb = *(const v4f*)(xs + d + 4);
    // 8 non-temporal b128 streaming loads (rows d..d+7, this thread's 4 cols)
    v4f e0 = __builtin_nontemporal_load(ep + (size_t)(d + 0) * (O_DIM / 4));
    v4f e1 = __builtin_nontemporal_load(ep + (size_t)(d + 1) * (O_DIM / 4));
    v4f e2 = __builtin_nontemporal_load(ep + (size_t)(d + 2) * (O_DIM / 4));
    v4f e3 = __builtin_nontemporal_load(ep + (size_t)(d + 3) * (O_DIM / 4));
    v4f e4 = __builtin_nontemporal_load(ep + (size_t)(d + 4) * (O_DIM / 4));
    v4f e5 = __builtin_nontemporal_load(ep + (size_t)(d + 5) * (O_DIM / 4));
    v4f e6 = __builtin_nontemporal_load(ep + (size_t)(d + 6) * (O_DIM / 4));
    v4f e7 = __builtin_nontemporal_load(ep + (size_t)(d + 7) * (O_DIM / 4));
    acc += xa.x * e0;
    acc += xa.y * e1;
    acc += xa.z * e2;
    acc += xa.w * e3;
    acc += xb.x * e4;
    acc += xb.y * e5;
    acc += xb.z * e6;
    acc += xb.w * e7;
  }

  v4f* op = (v4f*)(out + (size_t)p * O_DIM) + t;
  *op = *op + acc;   // base (+biases) written by kernel 1, same stream
}

extern "C" void kernel_launch(void* const* d_in, const int* in_sizes, int n_in,
                              void* d_out, int out_size, void* d_ws, size_t ws_size,
                              hipStream_t stream) {
  (void)in_sizes; (void)n_in; (void)out_size; (void)d_ws; (void)ws_size;
  const float* x        = (const float*)d_in[0];  // [P, D]
  const float* weight   = (const float*)d_in[1];  // [D, O]
  const float* bias     = (const float*)d_in[2];  // [O]
  const float* w_eps    = (const float*)d_in[3];  // [P, D, O]
  const float* b_eps    = (const float*)d_in[4];  // [P, O]
  float* out = (float*)d_out;

  // base GEMM + biases (WMMA, L2-resident operands)
  nes_base_wmma<<<dim3(O_DIM / 16, P_DIM / 16), dim3(32), 0, stream>>>(
      x, weight, bias, b_eps, out);

  // HBM-bound perturbation stream, accumulates onto out
  nes_eps_stream<<<dim3(P_DIM), dim3(128), 0, stream>>>(x, w_eps, out);
}